// ADCActivation_55465207660703
// MI455X (gfx1250) — compile-verified
//
#include <hip/hip_runtime.h>
#include <stdint.h>

// ---------------------------------------------------------------------------
// ADC activation:  out = 0.003125 * searchsorted(adc_char, clamp(x,0,7.9375), 'right')
// Memory-bound streaming op (411 MB @ 23.3 TB/s ~ 17.6 us floor).
// Strategy: exact power-of-two binning (x128) + per-block LDS LUT of
// cumulative threshold counts; rare boundary scan. TDM stages the table.
// ---------------------------------------------------------------------------

#define ADC_CLAMP_MAX 7.9375f     // 2^(7-4) - 2^-4
#define ADC_OUT_SCALE 0.003125f   // 0.05 / 16
#define ADC_NBINS     1018        // floor(128*7.9375)=1016 -> need cnt[0..1017]
#define ADC_MAXLEV    128         // thresholds padded to 128 (pad = +inf)

typedef float f32x4 __attribute__((ext_vector_type(4)));

#if defined(__HIP_DEVICE_COMPILE__) && __has_builtin(__builtin_amdgcn_tensor_load_to_lds)
#define ADC_USE_TDM 1
typedef unsigned int adc_u32x4 __attribute__((ext_vector_type(4)));
typedef int          adc_i32x4 __attribute__((ext_vector_type(4)));
typedef int          adc_i32x8 __attribute__((ext_vector_type(8)));
#endif

// Count of thresholds <= a, via bucket LUT. Exact: a*128 and t*128 are exact
// (power-of-two scale, values <= 1016), so bin comparisons are exact and only
// thresholds sharing a's bin need an explicit <= test.
__device__ __forceinline__ float adc_one(float xv,
                                         const float* __restrict__ adc_s,
                                         const unsigned char* __restrict__ cnt_s) {
    float a = fminf(fmaxf(xv, 0.0f), ADC_CLAMP_MAX);
    int b = (int)(a * 128.0f);          // exact floor, 0..1016
    int k = (int)cnt_s[b];              // #thresholds with bin < b (all <= a)
    int e = (int)cnt_s[b + 1];          // start of bins > b (all > a)
    int q = k;
    while (k < e) {                     // usually 0 iterations (density .125/bin)
        q += (adc_s[k] <= a) ? 1 : 0;
        ++k;
    }
    return (float)q * ADC_OUT_SCALE;
}

__global__ __launch_bounds__(256)
void adc_activation_kernel(const float* __restrict__ x,
                           const float* __restrict__ adc,
                           float* __restrict__ out,
                           int n, int nlev) {
    __shared__ float adc_s[ADC_MAXLEV];
    __shared__ unsigned char cnt_s[ADC_NBINS];

    const int tid = (int)threadIdx.x;

#ifdef ADC_USE_TDM
    // --- Stage threshold table into LDS with the Tensor Data Mover. ---------
    {
        uint64_t ga      = (uint64_t)(uintptr_t)(const void*)adc;
        uint32_t lds_off = (uint32_t)(uintptr_t)(void*)adc_s;  // low 32b = LDS offset
        uint32_t ulev    = (uint32_t)nlev;
        // D# group 0: count=1 | lds_addr | global_addr[56:0] | type=2
        adc_u32x4 g0 = { 1u,
                         lds_off,
                         (uint32_t)(ga & 0xFFFFFFFFu),
                         (uint32_t)((ga >> 32) & 0x01FFFFFFu) | (2u << 30) };
        // D# group 1: data_size=4B; tensor_dim0=nlev; tensor_dim1=1;
        //             tile_dim0=nlev; tile_dim1=1; dim0_stride=nlev
        adc_i32x8 g1 = { (int)(2u << 16),
                         (int)(ulev << 16),
                         (int)(1u << 16),
                         (int)(ulev << 16),
                         1,
                         (int)ulev,
                         0, 0 };
        adc_i32x4 gz = { 0, 0, 0, 0 };
        if (tid < 32) {                 // one wave issues the DMA (EXEC ignored)
#if __clang_major__ >= 23
            adc_i32x8 gz8 = { 0, 0, 0, 0, 0, 0, 0, 0 };
            __builtin_amdgcn_tensor_load_to_lds(g0, g1, gz, gz, gz8, 0);
#else
            __builtin_amdgcn_tensor_load_to_lds(g0, g1, gz, gz, 0);
#endif
            __builtin_amdgcn_s_wait_tensorcnt(0);
        }
    }
    for (int t = tid; t < ADC_MAXLEV; t += (int)blockDim.x)
        if (t >= nlev) adc_s[t] = __builtin_inff();     // +inf pad (disjoint from DMA)
#else
    for (int t = tid; t < ADC_MAXLEV; t += (int)blockDim.x)
        adc_s[t] = (t < nlev) ? adc[t] : __builtin_inff();
#endif
    __syncthreads();

    // --- Build cumulative-count LUT: cnt[b] = #{k : 128*adc[k] < b}. --------
    for (int b = tid; b < ADC_NBINS; b += (int)blockDim.x) {
        float fb = (float)b;
        int lo = 0;
#pragma unroll
        for (int s = 64; s >= 1; s >>= 1) {             // branchless upper bound
            int t = lo + s;
            lo = (adc_s[t - 1] * 128.0f < fb) ? t : lo; // exact compare
        }
        cnt_s[b] = (unsigned char)lo;
    }
    __syncthreads();

    // --- Streaming main loop: float4, non-temporal (no reuse). --------------
    const int gsize = (int)(blockDim.x * gridDim.x);
    const int gid   = (int)(blockIdx.x * blockDim.x) + tid;
    const int nvec  = n >> 2;
    const f32x4* __restrict__ x4 = (const f32x4*)x;
    f32x4* __restrict__ o4       = (f32x4*)out;

    for (int i = gid; i < nvec; i += gsize) {
        f32x4 v = __builtin_nontemporal_load(x4 + i);
        f32x4 r;
        r.x = adc_one(v.x, adc_s, cnt_s);
        r.y = adc_one(v.y, adc_s, cnt_s);
        r.z = adc_one(v.z, adc_s, cnt_s);
        r.w = adc_one(v.w, adc_s, cnt_s);
        __builtin_nontemporal_store(r, o4 + i);
    }
    for (int i = (nvec << 2) + gid; i < n; i += gsize)  // scalar tail
        out[i] = adc_one(x[i], adc_s, cnt_s);
}

extern "C" void kernel_launch(void* const* d_in, const int* in_sizes, int n_in,
                              void* d_out, int out_size, void* d_ws, size_t ws_size,
                              hipStream_t stream) {
    (void)n_in; (void)out_size; (void)d_ws; (void)ws_size;
    const float* x   = (const float*)d_in[0];
    const float* adc = (const float*)d_in[1];
    float* out       = (float*)d_out;
    const int n    = in_sizes[0];
    const int nlev = in_sizes[1];   // 127 in the reference

    const int threads = 256;        // 8 wave32s / block
    int blocks = 6144;              // grid-stride: ~8 float4 iters/thread at n=51.4M
    int need = ((n >> 2) + threads - 1) / threads;
    if (need < 1) need = 1;
    if (blocks > need) blocks = need;

    hipLaunchKernelGGL(adc_activation_kernel, dim3(blocks), dim3(threads), 0, stream,
                       x, adc, out, n, nlev);
}